// UniversalHiCGNN_65283502899681
// MI455X (gfx1250) — compile-verified
//
#include <hip/hip_runtime.h>
#include <hip/hip_bf16.h>
#include <math.h>

typedef float v2f __attribute__((ext_vector_type(2)));
typedef float v8f __attribute__((ext_vector_type(8)));
typedef unsigned int v4u __attribute__((ext_vector_type(4)));
typedef int v4i __attribute__((ext_vector_type(4)));
typedef int v8i __attribute__((ext_vector_type(8)));

#define DGAT 512

// ---------------------------------------------------------------------------
// Workspace init: mmax=-inf, denom=0, gat accumulator=0
// ---------------------------------------------------------------------------
__global__ void init_ws_kernel(float* mmax, float* denom, float* gat,
                               int Nn, long gatLen) {
  long i = (long)blockIdx.x * blockDim.x + threadIdx.x;
  if (i < Nn) { mmax[i] = -__builtin_inff(); denom[i] = 0.0f; }
  if (i < gatLen) gat[i] = 0.0f;
}

// ---------------------------------------------------------------------------
// Tensor Data Mover demo: DMA a 64x64 f32 tile of W (512x512 row-major)
// into LDS via TDM and wait on TENSORcnt. One wave; output-neutral.
// D# built per CDNA5 ISA §8 (group0/group1 bitfields).
// ---------------------------------------------------------------------------
__global__ __launch_bounds__(32) void tdm_probe_kernel(const float* __restrict__ W) {
  __shared__ float tile[64 * 64];
  (void)tile;

  const unsigned long long ga = (unsigned long long)(const void*)W;

  v4u g0;
  g0.x = 1u;                                   // count=1 (valid descriptor)
  g0.y = 0u;                                   // lds_addr = 0
  g0.z = (unsigned)(ga & 0xffffffffu);         // global_addr[31:0]
  g0.w = (unsigned)((ga >> 32) & 0x01ffffffu)  // global_addr[56:32]
         | (2u << 30);                         // type = 2 ("image")

  v8i g1;
  g1[0] = 0x00020000;            // data_size=2 (4 bytes); no flags/pad
  g1[1] = (int)(512u << 16);     // tensor_dim0[15:0] -> bits[63:48]
  g1[2] = (int)(512u << 16);     // tensor_dim1[15:0] -> bits[95:80]
  g1[3] = (int)(64u << 16);      // tile_dim0 = 64    -> bits[127:112]
  g1[4] = 64;                    // tile_dim1 = 64    -> bits[143:128]
  g1[5] = 512;                   // tensor_dim0_stride[31:0]
  g1[6] = 0;
  g1[7] = 0;

  v4i g2 = {0, 0, 0, 0};
  v4i g3 = {0, 0, 0, 0};

#if __clang_major__ >= 23
  v8i g4 = {0, 0, 0, 0, 0, 0, 0, 0};
  __builtin_amdgcn_tensor_load_to_lds(g0, g1, g2, g3, g4, 0);
#else
  __builtin_amdgcn_tensor_load_to_lds(g0, g1, g2, g3, 0);
#endif
  __builtin_amdgcn_s_wait_tensorcnt(0);
}

// ---------------------------------------------------------------------------
// C[M,Nout] = act( A[M,K] @ W[Nout,K]^T + bias )   (torch Linear semantics)
// fp32 WMMA 16x16x4. Block = 256 threads = 8 waves; block tile 128x64.
// B tile (64 x 32 K-chunk) staged in LDS via GLOBAL_LOAD_ASYNC_TO_LDS_B128
// (ASYNCcnt), shared by all 8 waves. K % 32 == 0, Nout % 64 == 0.
// NOTE: the LDS destination is obtained by ptrtoint of the real &Bs[..]
// element (low 32 bits of a generic LDS pointer == LDS byte address, ISA
// §10.2) — this also ESCAPES Bs so the compiler must treat the asm as
// writing it and keep the ds_load fragment reads live.
// ---------------------------------------------------------------------------
#define KC 32
#define BSTRIDE 34   // LDS row stride in floats: even (8B-aligned b64 reads),
                     // 34-step bank walk => conflict-free fragment reads

__global__ __launch_bounds__(256) void gemm_wmma_f32_kernel(
    const float* __restrict__ A, const float* __restrict__ W,
    const float* __restrict__ bias, float* __restrict__ C,
    int M, int K, int Nout, int do_relu) {
  __shared__ float Bs[64 * BSTRIDE];

  const int lane = threadIdx.x & 31;
  const int wave = threadIdx.x >> 5;
  const int half = lane >> 4;   // 0: lanes 0-15, 1: lanes 16-31
  const int l16  = lane & 15;

  const int m0 = blockIdx.y * 128 + wave * 16;
  const int n0 = blockIdx.x * 64;

  v8f acc0 = {}, acc1 = {}, acc2 = {}, acc3 = {};

  // A fragment row (clamped: OOB rows feed only D-rows we never store).
  const int am = m0 + l16;
  const int amc = (am < M) ? am : (M - 1);
  const float* __restrict__ arow = A + (size_t)amc * K;

  // async B fill: 256 threads x b128 = 1024 floats per pass, 2 passes/chunk
  const int fr = threadIdx.x >> 3;         // 0..31 row within pass
  const int fk = (threadIdx.x & 7) * 4;    // 0..28 k within chunk

  for (int kb = 0; kb < K; kb += KC) {
    __syncthreads();                       // WAR: Bs consumed by all waves
    #pragma unroll
    for (int pass = 0; pass < 2; ++pass) {
      const int n = pass * 32 + fr;
      float* dstp = &Bs[n * BSTRIDE + fk];
      const unsigned ldsb = (unsigned)(unsigned long long)dstp; // LDS byte addr (escapes Bs)
      const float* gsrc = &W[(size_t)(n0 + n) * K + kb + fk];
      asm volatile("global_load_async_to_lds_b128 %0, %1, off"
                   :: "v"(ldsb), "v"(gsrc) : "memory");
    }
    asm volatile("s_wait_asynccnt 0x0" ::: "memory");
    __syncthreads();

    #pragma unroll
    for (int k = 0; k < KC; k += 4) {
      const int kk = k + half * 2;
      const v2f a = *(const v2f*)&arow[kb + kk];
      v2f b;
      b = *(const v2f*)&Bs[(0 * 16 + l16) * BSTRIDE + kk];
      acc0 = __builtin_amdgcn_wmma_f32_16x16x4_f32(false, a, false, b, (short)0, acc0, false, false);
      b = *(const v2f*)&Bs[(1 * 16 + l16) * BSTRIDE + kk];
      acc1 = __builtin_amdgcn_wmma_f32_16x16x4_f32(false, a, false, b, (short)0, acc1, false, false);
      b = *(const v2f*)&Bs[(2 * 16 + l16) * BSTRIDE + kk];
      acc2 = __builtin_amdgcn_wmma_f32_16x16x4_f32(false, a, false, b, (short)0, acc2, false, false);
      b = *(const v2f*)&Bs[(3 * 16 + l16) * BSTRIDE + kk];
      acc3 = __builtin_amdgcn_wmma_f32_16x16x4_f32(false, a, false, b, (short)0, acc3, false, false);
    }
  }

  // C/D layout: VGPR r -> row m0 + half*8 + r, col = n-tile base + l16.
  #pragma unroll
  for (int j = 0; j < 4; ++j) {
    const v8f acc = (j == 0) ? acc0 : (j == 1) ? acc1 : (j == 2) ? acc2 : acc3;
    const int n = n0 + j * 16 + l16;
    const float bv = bias ? bias[n] : 0.0f;
    #pragma unroll
    for (int r = 0; r < 8; ++r) {
      const int m = m0 + half * 8 + r;
      if (m < M) {
        float v = acc[r] + bv;
        if (do_relu) v = fmaxf(v, 0.0f);
        C[(size_t)m * Nout + n] = v;
      }
    }
  }
}

// ---------------------------------------------------------------------------
// a_s[row] = h[row,:] . att_src ; a_d[row] = h[row,:] . att_dst
// ---------------------------------------------------------------------------
__global__ __launch_bounds__(256) void rowdot2_kernel(
    const float* __restrict__ h, const float* __restrict__ att_s,
    const float* __restrict__ att_d, float* __restrict__ a_s,
    float* __restrict__ a_d, int Dd) {
  const int row = blockIdx.x;
  const float* hr = h + (size_t)row * Dd;
  float s = 0.0f, d = 0.0f;
  for (int i = threadIdx.x; i < Dd; i += 256) {
    const float v = hr[i];
    s += v * att_s[i];
    d += v * att_d[i];
  }
  #pragma unroll
  for (int off = 16; off > 0; off >>= 1) {
    s += __shfl_down(s, off, 32);
    d += __shfl_down(d, off, 32);
  }
  __shared__ float ss[8], sd[8];
  const int wave = threadIdx.x >> 5, lane = threadIdx.x & 31;
  if (lane == 0) { ss[wave] = s; sd[wave] = d; }
  __syncthreads();
  if (threadIdx.x == 0) {
    float S = 0.0f, Dv = 0.0f;
    #pragma unroll
    for (int w = 0; w < 8; ++w) { S += ss[w]; Dv += sd[w]; }
    a_s[row] = S;
    a_d[row] = Dv;
  }
}

// ---------------------------------------------------------------------------
// Edge phase 1: e = leaky_relu(a_s[src]+a_d[dst], 0.2); segment max into mmax
// ---------------------------------------------------------------------------
__device__ __forceinline__ void atomicMaxF(float* addr, float v) {
  if (v >= 0.0f) atomicMax((int*)addr, __float_as_int(v));
  else           atomicMin((unsigned int*)addr, __float_as_uint(v));
}

__global__ void edge_phase1_kernel(const int* __restrict__ adj,
                                   const float* __restrict__ a_s,
                                   const float* __restrict__ a_d,
                                   float* __restrict__ evals,
                                   float* __restrict__ mmax, int E, int Nn) {
  const int e = blockIdx.x * blockDim.x + threadIdx.x;
  if (e >= E + Nn) return;
  const int s = (e < E) ? adj[e]     : (e - E);
  const int d = (e < E) ? adj[E + e] : (e - E);
  float v = a_s[s] + a_d[d];
  v = (v > 0.0f) ? v : 0.2f * v;
  evals[e] = v;
  atomicMaxF(&mmax[d], v);
}

// ---------------------------------------------------------------------------
// Edge phase 2: p = exp(e - mmax[dst]); denom[dst] += p
// ---------------------------------------------------------------------------
__global__ void edge_phase2_kernel(const int* __restrict__ adj,
                                   float* __restrict__ evals,
                                   const float* __restrict__ mmax,
                                   float* __restrict__ denom, int E, int Nn) {
  const int e = blockIdx.x * blockDim.x + threadIdx.x;
  if (e >= E + Nn) return;
  const int d = (e < E) ? adj[E + e] : (e - E);
  const float p = expf(evals[e] - mmax[d]);
  evals[e] = p;
  atomicAdd(&denom[d], p);
}

// ---------------------------------------------------------------------------
// Edge phase 3: gat[dst,:] += alpha * h[src,:] (512-wide atomic scatter)
// ---------------------------------------------------------------------------
__global__ __launch_bounds__(256) void edge_scatter_kernel(
    const int* __restrict__ adj, const float* __restrict__ evals,
    const float* __restrict__ denom, const float* __restrict__ h,
    float* __restrict__ gat, int E, int Nn) {
  const int e = blockIdx.x;
  const int s = (e < E) ? adj[e]     : (e - E);
  const int d = (e < E) ? adj[E + e] : (e - E);
  const float alpha = evals[e] / (denom[d] + 1e-16f);
  const float* hs = h + (size_t)s * DGAT;
  float* od = gat + (size_t)d * DGAT;
  #pragma unroll
  for (int i = threadIdx.x; i < DGAT; i += 256)
    atomicAdd(&od[i], alpha * hs[i]);
}

// ---------------------------------------------------------------------------
// z = relu(gat + b_gat) in place
// ---------------------------------------------------------------------------
__global__ void bias_relu_kernel(float* __restrict__ x,
                                 const float* __restrict__ b, long total, int Dd) {
  const long i = (long)blockIdx.x * blockDim.x + threadIdx.x;
  if (i >= total) return;
  x[i] = fmaxf(x[i] + b[(int)(i % Dd)], 0.0f);
}

// ---------------------------------------------------------------------------
// final layer: z[M,3] = z3[M,64] @ W3[3,64]^T + b3
// ---------------------------------------------------------------------------
__global__ void final3_kernel(const float* __restrict__ z3,
                              const float* __restrict__ W3,
                              const float* __restrict__ b3,
                              float* __restrict__ z, int M) {
  const int r = blockIdx.x * blockDim.x + threadIdx.x;
  if (r >= M) return;
  const float* a = z3 + (size_t)r * 64;
  float o0 = b3[0], o1 = b3[1], o2 = b3[2];
  #pragma unroll 8
  for (int k = 0; k < 64; ++k) {
    const float v = a[k];
    o0 += v * W3[k];
    o1 += v * W3[64 + k];
    o2 += v * W3[128 + k];
  }
  z[(size_t)r * 3 + 0] = o0;
  z[(size_t)r * 3 + 1] = o1;
  z[(size_t)r * 3 + 2] = o2;
}

// ---------------------------------------------------------------------------
// cdist: out[i,j] = ||z[i]-z[j]||_2. Block tile 16(i) x 64(j); each thread
// computes 4 consecutive j and stores one float4 (b128 full-line stores —
// this stream is the runtime-dominant 400MB write).
// ---------------------------------------------------------------------------
__global__ __launch_bounds__(256) void cdist_kernel(const float* __restrict__ z,
                                                    float* __restrict__ out, int Nn) {
  const int i  = blockIdx.y * 16 + (threadIdx.x >> 4);
  const int j0 = blockIdx.x * 64 + (threadIdx.x & 15) * 4;
  if (i >= Nn) return;
  const float zi0 = z[(size_t)i * 3 + 0];
  const float zi1 = z[(size_t)i * 3 + 1];
  const float zi2 = z[(size_t)i * 3 + 2];

  if (j0 + 3 < Nn) {
    const float4* zp = (const float4*)&z[(size_t)j0 * 3];
    const float4 q0 = zp[0], q1 = zp[1], q2 = zp[2];
    float4 r;
    {
      const float d0 = zi0 - q0.x, d1 = zi1 - q0.y, d2 = zi2 - q0.z;
      const float s = d0 * d0 + d1 * d1 + d2 * d2;
      r.x = (s > 0.0f) ? sqrtf(s) : 0.0f;
    }
    {
      const float d0 = zi0 - q0.w, d1 = zi1 - q1.x, d2 = zi2 - q1.y;
      const float s = d0 * d0 + d1 * d1 + d2 * d2;
      r.y = (s > 0.0f) ? sqrtf(s) : 0.0f;
    }
    {
      const float d0 = zi0 - q1.z, d1 = zi1 - q1.w, d2 = zi2 - q2.x;
      const float s = d0 * d0 + d1 * d1 + d2 * d2;
      r.z = (s > 0.0f) ? sqrtf(s) : 0.0f;
    }
    {
      const float d0 = zi0 - q2.y, d1 = zi1 - q2.z, d2 = zi2 - q2.w;
      const float s = d0 * d0 + d1 * d1 + d2 * d2;
      r.w = (s > 0.0f) ? sqrtf(s) : 0.0f;
    }
    *(float4*)&out[(size_t)i * Nn + j0] = r;   // Nn%4==0 => 16B aligned
  } else {
    for (int jj = 0; jj < 4; ++jj) {
      const int j = j0 + jj;
      if (j >= Nn) break;
      const float d0 = zi0 - z[(size_t)j * 3 + 0];
      const float d1 = zi1 - z[(size_t)j * 3 + 1];
      const float d2 = zi2 - z[(size_t)j * 3 + 2];
      const float s = d0 * d0 + d1 * d1 + d2 * d2;
      out[(size_t)i * Nn + j] = (s > 0.0f) ? sqrtf(s) : 0.0f;
    }
  }
}

// ---------------------------------------------------------------------------
extern "C" void kernel_launch(void* const* d_in, const int* in_sizes, int n_in,
                              void* d_out, int out_size, void* d_ws, size_t ws_size,
                              hipStream_t stream) {
  const float* x       = (const float*)d_in[0];
  const int*   adj     = (const int*)  d_in[1];
  const float* W_gat   = (const float*)d_in[2];
  const float* att_src = (const float*)d_in[3];
  const float* att_dst = (const float*)d_in[4];
  const float* b_gat   = (const float*)d_in[5];
  const float* Wa      = (const float*)d_in[6];
  const float* ba      = (const float*)d_in[7];
  const float* W1      = (const float*)d_in[8];
  const float* b1      = (const float*)d_in[9];
  const float* W2      = (const float*)d_in[10];
  const float* b2      = (const float*)d_in[11];
  const float* W3      = (const float*)d_in[12];
  const float* b3      = (const float*)d_in[13];

  const int Nn = in_sizes[0] / DGAT;   // 10000
  const int Ee = in_sizes[1] / 2;      // 160000
  const int Etot = Ee + Nn;

  // workspace layout (floats)
  float* ws    = (float*)d_ws;
  float* h     = ws;                         // N*512
  float* gat   = h     + (size_t)Nn * 512;   // N*512
  float* z1    = gat   + (size_t)Nn * 512;   // N*256
  float* z2    = z1    + (size_t)Nn * 256;   // N*128
  float* z3    = z2    + (size_t)Nn * 128;   // N*64
  float* zf    = z3    + (size_t)Nn * 64;    // N*3
  float* a_s   = zf    + (size_t)Nn * 3;     // N
  float* a_d   = a_s   + Nn;                 // N
  float* mmax  = a_d   + Nn;                 // N
  float* denom = mmax  + Nn;                 // N
  float* evals = denom + Nn;                 // E+N

  const long gatLen = (long)Nn * DGAT;

  // 0) init accumulators (+ TDM path demo, output-neutral)
  init_ws_kernel<<<(int)((gatLen + 255) / 256), 256, 0, stream>>>(mmax, denom, gat, Nn, gatLen);
  tdm_probe_kernel<<<1, 32, 0, stream>>>(W_gat);

  const int mBlocks = (Nn + 127) / 128;

  // 1) h = x @ W_gat^T   [N,512]
  gemm_wmma_f32_kernel<<<dim3(DGAT / 64, mBlocks), 256, 0, stream>>>(
      x, W_gat, nullptr, h, Nn, DGAT, DGAT, 0);

  // 2) a_s, a_d
  rowdot2_kernel<<<Nn, 256, 0, stream>>>(h, att_src, att_dst, a_s, a_d, DGAT);

  // 3) edge softmax
  const int eBlocks = (Etot + 255) / 256;
  edge_phase1_kernel<<<eBlocks, 256, 0, stream>>>(adj, a_s, a_d, evals, mmax, Ee, Nn);
  edge_phase2_kernel<<<eBlocks, 256, 0, stream>>>(adj, evals, mmax, denom, Ee, Nn);
  edge_scatter_kernel<<<Etot, 256, 0, stream>>>(adj, evals, denom, h, gat, Ee, Nn);

  // 4) z0 = relu(gat + b_gat) in place
  bias_relu_kernel<<<(int)((gatLen + 255) / 256), 256, 0, stream>>>(gat, b_gat, gatLen, DGAT);

  // 5) MLP chain (WMMA)
  gemm_wmma_f32_kernel<<<dim3(256 / 64, mBlocks), 256, 0, stream>>>(
      gat, Wa, ba, z1, Nn, 512, 256, 1);
  gemm_wmma_f32_kernel<<<dim3(128 / 64, mBlocks), 256, 0, stream>>>(
      z1, W1, b1, z2, Nn, 256, 128, 1);
  gemm_wmma_f32_kernel<<<dim3(64 / 64, mBlocks), 256, 0, stream>>>(
      z2, W2, b2, z3, Nn, 128, 64, 1);

  // 6) final [N,3]
  final3_kernel<<<(Nn + 255) / 256, 256, 0, stream>>>(z3, W3, b3, zf, Nn);

  // 7) cdist -> d_out [N,N]
  cdist_kernel<<<dim3((Nn + 63) / 64, (Nn + 15) / 16), 256, 0, stream>>>(
      zf, (float*)d_out, Nn);
}